// MICRO_52304111730856
// MI455X (gfx1250) — compile-verified
//
#include <hip/hip_runtime.h>
#include <hip/hip_bf16.h>
#include <math.h>

typedef __attribute__((ext_vector_type(2))) float v2f;
typedef __attribute__((ext_vector_type(4))) float v4f;
typedef __attribute__((ext_vector_type(8))) float v8f;
typedef __attribute__((ext_vector_type(4))) int   v4i;

#define NUSER 24000
#define NITEM 8000
#define DIM   64
#define VDIM  2048
#define TDIM  384
#define TOPKN 10
#define NE2   800000      // 2 * N_INTER
#define NTOT  (NUSER + NITEM)
#define NEG_INF (-3.402823466e38f)

#if __has_builtin(__builtin_amdgcn_global_load_async_to_lds_b128)
#define HAVE_GFX1250_ASYNC 1
#endif

#ifdef HAVE_GFX1250_ASYNC
typedef __attribute__((address_space(1))) v4i* gv4i_ptr;   // global int4*
typedef __attribute__((address_space(3))) v4i* lv4i_ptr;   // LDS int4*
#endif

// 16-byte global -> LDS copy. Async (ASYNCcnt-tracked) when the gfx1250
// builtin exists, synchronous v4f copy otherwise.
__device__ __forceinline__ void stage16(const float* __restrict__ g, float* l) {
#ifdef HAVE_GFX1250_ASYNC
  __builtin_amdgcn_global_load_async_to_lds_b128(
      (gv4i_ptr)(unsigned long long)(const void*)g,
      (lv4i_ptr)(unsigned int)(unsigned long long)(const void*)l,
      0, 0);
#else
  *(v4f*)l = *(const v4f*)g;
#endif
}

__device__ __forceinline__ void wait_async_lds() {
#ifdef HAVE_GFX1250_ASYNC
#if __has_builtin(__builtin_amdgcn_s_wait_asynccnt)
  __builtin_amdgcn_s_wait_asynccnt(0);
#else
  asm volatile("s_wait_asynccnt 0x0" ::: "memory");
#endif
#endif
}

__device__ __forceinline__ float wave_reduce_sum(float v) {
  #pragma unroll
  for (int m = 16; m > 0; m >>= 1) v += __shfl_xor(v, m, 32);
  return v;
}

// ---------------------------------------------------------------------------
// K1: C[M x 64] = A[M x K] @ W[K x 64] + bias   via v_wmma_f32_16x16x4_f32
// one wave per 16-row tile; 4 column tiles of 16; K stepped by 4.
// ---------------------------------------------------------------------------
__global__ void proj_gemm_wmma(const float* __restrict__ A,
                               const float* __restrict__ W,
                               const float* __restrict__ bias,
                               float* __restrict__ C, int M, int K) {
  int wave = (blockIdx.x * blockDim.x + threadIdx.x) >> 5;
  int lane = threadIdx.x & 31;
  int ntiles = M >> 4;
  if (wave >= ntiles) return;              // whole-wave exit: EXEC stays all-1s
  int i0 = wave << 4;
  int half = lane >> 4;                    // 0: lanes 0-15, 1: lanes 16-31
  int n = lane & 15;                       // N index (and M index for A)
  v8f acc0 = {}, acc1 = {}, acc2 = {}, acc3 = {};
  const float* arow = A + (size_t)(i0 + n) * K;
  for (int k0 = 0; k0 < K; k0 += 4) {
    int ka = k0 + 2 * half;                // A: K = vgpr + 2*half
    v2f a = *(const v2f*)(arow + ka);
    const float* w0 = W + (size_t)ka * DIM;
    const float* w1 = W + (size_t)(ka + 1) * DIM;
    v2f b0; b0.x = w0[n];      b0.y = w1[n];
    v2f b1; b1.x = w0[16 + n]; b1.y = w1[16 + n];
    v2f b2; b2.x = w0[32 + n]; b2.y = w1[32 + n];
    v2f b3; b3.x = w0[48 + n]; b3.y = w1[48 + n];
    acc0 = __builtin_amdgcn_wmma_f32_16x16x4_f32(false, a, false, b0, (short)0, acc0, false, false);
    acc1 = __builtin_amdgcn_wmma_f32_16x16x4_f32(false, a, false, b1, (short)0, acc1, false, false);
    acc2 = __builtin_amdgcn_wmma_f32_16x16x4_f32(false, a, false, b2, (short)0, acc2, false, false);
    acc3 = __builtin_amdgcn_wmma_f32_16x16x4_f32(false, a, false, b3, (short)0, acc3, false, false);
  }
  #pragma unroll
  for (int r = 0; r < 8; ++r) {
    int row = i0 + r + 8 * half;           // C/D layout: M = vgpr + 8*half
    float* crow = C + (size_t)row * DIM;
    crow[n]      = acc0[r] + bias[n];
    crow[16 + n] = acc1[r] + bias[16 + n];
    crow[32 + n] = acc2[r] + bias[32 + n];
    crow[48 + n] = acc3[r] + bias[48 + n];
  }
}

// ---------------------------------------------------------------------------
// K2: row L2-normalize (one wave per 64-dim row)
// ---------------------------------------------------------------------------
__global__ void row_normalize(const float* __restrict__ x, float* __restrict__ xn, int M) {
  int wave = (blockIdx.x * blockDim.x + threadIdx.x) >> 5;
  int lane = threadIdx.x & 31;
  if (wave >= M) return;
  v2f v = *(const v2f*)(x + (size_t)wave * DIM + lane * 2);
  float ss = wave_reduce_sum(v.x * v.x + v.y * v.y);
  float inv = __frsqrt_rn(ss);
  v2f o; o.x = v.x * inv; o.y = v.y * inv;
  *(v2f*)(xn + (size_t)wave * DIM + lane * 2) = o;
}

__device__ __forceinline__ void topk_insert(float (&tv)[TOPKN], int (&tc)[TOPKN],
                                            float v, int c) {
  if (v <= tv[TOPKN - 1]) return;          // common fast path
  #pragma unroll
  for (int s = 0; s < TOPKN; ++s) {
    bool gt = v > tv[s];
    float nv = gt ? v : tv[s]; float sv = gt ? tv[s] : v;
    int   ni = gt ? c : tc[s]; int   si = gt ? tc[s] : c;
    tv[s] = nv; tc[s] = ni; v = sv; c = si;
  }
}

// ---------------------------------------------------------------------------
// K3: fused  sim = xn @ xn^T  (16-row stripe per wave, K=64)  +  running top-10.
// A-panel (16x64 f32) lives in 32 VGPRs for the whole sweep.  The 16x64 B-panel
// of each column tile is staged once per block into LDS (double-buffered) with
// GLOBAL_LOAD_ASYNC_TO_LDS_B128 and shared by all 8 waves -> 8x less L2
// traffic, staging overlapped with the 16 v_wmma per tile.  C tiles spill to
// per-wave LDS; lane/lane^16 split the 16 columns of a row and keep private
// sorted top-10 lists, merged at the end via __shfl.
// ---------------------------------------------------------------------------
__global__ void sim_topk_wmma(const float* __restrict__ xn,
                              float* __restrict__ topv, int* __restrict__ topc) {
  __shared__ float bpanel[2][16 * DIM];    // 2 x 4KB double buffer
  __shared__ float tile[8 * 256];          // 1KB per wave C-tile spill
  const int t    = threadIdx.x;            // 0..255
  const int wib  = t >> 5;
  const int lane = t & 31;
  const int NT   = NITEM >> 4;             // 500 column tiles / stripes
  int stripe = blockIdx.x * 8 + wib;
  bool active = stripe < NT;
  if (!active) stripe = NT - 1;            // tail waves compute redundantly,
  int i0 = stripe << 4;                    // keeping barrier counts uniform
  int half = lane >> 4;
  int n = lane & 15;

  // resident A panel: rows i0..i0+15, all K
  v2f aK[16];
  const float* arow = xn + (size_t)(i0 + n) * DIM;
  #pragma unroll
  for (int kk = 0; kk < 16; ++kk) aK[kk] = *(const v2f*)(arow + kk * 4 + 2 * half);

  float tv[TOPKN]; int tc[TOPKN];
  #pragma unroll
  for (int s = 0; s < TOPKN; ++s) { tv[s] = NEG_INF; tc[s] = 0; }

  // prologue: stage tile 0 into buffer 0 (256 threads x 16B = 4KB)
  stage16(xn + (size_t)t * 4, &bpanel[0][t * 4]);
  wait_async_lds();
  __syncthreads();

  float* mytile = tile + wib * 256;
  for (int jt = 0; jt < NT; ++jt) {
    int cur = jt & 1;
    if (jt + 1 < NT)                       // kick next panel while computing
      stage16(xn + ((size_t)(jt + 1) * 16) * DIM + t * 4, &bpanel[cur ^ 1][t * 4]);

    int j0 = jt << 4;
    v8f acc = {};
    const float* bbase = &bpanel[cur][n * DIM + 2 * half];
    #pragma unroll
    for (int kk = 0; kk < 16; ++kk) {
      v2f b = *(const v2f*)(bbase + kk * 4);
      acc = __builtin_amdgcn_wmma_f32_16x16x4_f32(false, aK[kk], false, b,
                                                  (short)0, acc, false, false);
    }
    #pragma unroll
    for (int r = 0; r < 8; ++r) mytile[(r + 8 * half) * 16 + n] = acc[r];
    // lane owns row (lane&15); low half scans cols 0-7, high half cols 8-15
    const float* rowv = mytile + n * 16 + half * 8;
    #pragma unroll
    for (int c = 0; c < 8; ++c)
      topk_insert(tv, tc, rowv[c], j0 + half * 8 + c);

    wait_async_lds();                      // next panel resident
    __syncthreads();                       // everyone done with cur panel/tile
  }

  // merge the two half-lists of each row (partner = lane ^ 16)
  #pragma unroll
  for (int s = 0; s < TOPKN; ++s) {
    float pv = __shfl(tv[s], lane | 16, 32);
    int   pc = __shfl(tc[s], lane | 16, 32);
    if (lane < 16) topk_insert(tv, tc, pv, pc);
  }
  if (active && lane < 16) {
    float* ov = topv + (size_t)(i0 + lane) * TOPKN;
    int*   oc = topc + (size_t)(i0 + lane) * TOPKN;
    #pragma unroll
    for (int s = 0; s < TOPKN; ++s) { ov[s] = tv[s]; oc[s] = tc[s]; }
  }
}

// ---------------------------------------------------------------------------
// K4: dis[row] = deg>0 ? deg^-0.5 : 0,  deg = sum of row's top-10 weights
// ---------------------------------------------------------------------------
__global__ void compute_dis(const float* __restrict__ topv, float* __restrict__ dis, int M) {
  int i = blockIdx.x * blockDim.x + threadIdx.x;
  if (i >= M) return;
  float d = 0.f;
  #pragma unroll
  for (int s = 0; s < TOPKN; ++s) d += topv[(size_t)i * TOPKN + s];
  dis[i] = d > 0.f ? __frsqrt_rn(d) : 0.f;
}

// ---------------------------------------------------------------------------
// K5: out[i] = sum_k 0.1*(dis[i]*v*dis[c]) * item_emb[c]  +  0.9*ow * item_emb[oc]
//     one wave per row, 2 dims per lane
// ---------------------------------------------------------------------------
__global__ void spmm_knn(const float* __restrict__ item_emb,
                         const float* __restrict__ topv, const int* __restrict__ topc,
                         const float* __restrict__ dis,
                         const float* __restrict__ ow, const int* __restrict__ oc,
                         float* __restrict__ out) {
  int row  = (blockIdx.x * blockDim.x + threadIdx.x) >> 5;
  int lane = threadIdx.x & 31;
  if (row >= NITEM) return;
  int d0 = lane * 2;
  float ax = 0.f, ay = 0.f;
  float dr = dis[row];
  #pragma unroll
  for (int k = 0; k < TOPKN; ++k) {
    float v = topv[(size_t)row * TOPKN + k];
    int   c = topc[(size_t)row * TOPKN + k];
    float w = 0.1f * dr * v * dis[c];
    v2f e = *(const v2f*)(item_emb + (size_t)c * DIM + d0);
    ax += w * e.x; ay += w * e.y;
    float w2 = 0.9f * ow[(size_t)row * TOPKN + k];
    int   c2 = oc[(size_t)row * TOPKN + k];
    v2f e2 = *(const v2f*)(item_emb + (size_t)c2 * DIM + d0);
    ax += w2 * e2.x; ay += w2 * e2.y;
  }
  v2f o; o.x = ax; o.y = ay;
  *(v2f*)(out + (size_t)row * DIM + d0) = o;
}

// ---------------------------------------------------------------------------
// K6: attention gate + fused h row-normalize.  Wq1 cached in LDS.
// ---------------------------------------------------------------------------
__global__ void attention_h(const float* __restrict__ img, const float* __restrict__ txt,
                            const float* __restrict__ Wq1, const float* __restrict__ bq1,
                            const float* __restrict__ wq2, float* __restrict__ h_norm) {
  __shared__ float sW[DIM * DIM];
  __shared__ float sb[DIM], sw2[DIM];
  for (int t = threadIdx.x; t < DIM * DIM; t += blockDim.x) sW[t] = Wq1[t];
  if (threadIdx.x < DIM) { sb[threadIdx.x] = bq1[threadIdx.x]; sw2[threadIdx.x] = wq2[threadIdx.x]; }
  __syncthreads();
  int nwaves = gridDim.x * (blockDim.x >> 5);
  int wave = blockIdx.x * (blockDim.x >> 5) + (threadIdx.x >> 5);
  int lane = threadIdx.x & 31;
  int d0 = lane * 2;
  for (int row = wave; row < NITEM; row += nwaves) {
    const float* xi = img + (size_t)row * DIM;
    const float* xt = txt + (size_t)row * DIM;
    float si0 = sb[d0], si1 = sb[d0 + 1], st0 = sb[d0], st1 = sb[d0 + 1];
    for (int k = 0; k < DIM; ++k) {
      float w0 = sW[k * DIM + d0], w1 = sW[k * DIM + d0 + 1];
      float xik = xi[k], xtk = xt[k];     // same addr across lanes -> broadcast
      si0 += xik * w0; si1 += xik * w1;
      st0 += xtk * w0; st1 += xtk * w1;
    }
    float pi = tanhf(si0) * sw2[d0] + tanhf(si1) * sw2[d0 + 1];
    float pt = tanhf(st0) * sw2[d0] + tanhf(st1) * sw2[d0 + 1];
    float att_i = wave_reduce_sum(pi);
    float att_t = wave_reduce_sum(pt);
    float mx = fmaxf(att_i, att_t);
    float ei = __expf(att_i - mx), et = __expf(att_t - mx);
    float inv = 1.f / (ei + et);
    float wi = ei * inv, wt = et * inv;
    v2f a = *(const v2f*)(xi + d0);
    v2f b = *(const v2f*)(xt + d0);
    float hx = wi * a.x + wt * b.x, hy = wi * a.y + wt * b.y;
    float ss = wave_reduce_sum(hx * hx + hy * hy);
    float nrm = fmaxf(__fsqrt_rn(ss), 1e-12f);
    v2f o; o.x = hx / nrm; o.y = hy / nrm;
    *(v2f*)(h_norm + (size_t)row * DIM + d0) = o;
  }
}

// ---------------------------------------------------------------------------
// UI graph kernels
// ---------------------------------------------------------------------------
__global__ void zero_f32(float* __restrict__ p, size_t n) {
  for (size_t i = blockIdx.x * (size_t)blockDim.x + threadIdx.x; i < n;
       i += (size_t)gridDim.x * blockDim.x) p[i] = 0.f;
}

__global__ void build_ego(const float* __restrict__ u, const float* __restrict__ it,
                          float* __restrict__ ego) {
  size_t i = blockIdx.x * (size_t)blockDim.x + threadIdx.x;
  if (i >= (size_t)NTOT * DIM) return;
  ego[i] = (i < (size_t)NUSER * DIM) ? u[i] : it[i - (size_t)NUSER * DIM];
}

__global__ void deg_count(const int* __restrict__ rows, float* __restrict__ deg) {
  int e = blockIdx.x * blockDim.x + threadIdx.x;
  if (e >= NE2) return;
  atomicAdd(&deg[rows[e]], 1.0f);
}

__global__ void dis_from_deg(const float* __restrict__ deg, float* __restrict__ dis, int n) {
  int i = blockIdx.x * blockDim.x + threadIdx.x;
  if (i >= n) return;
  float d = deg[i];
  dis[i] = d > 0.f ? __frsqrt_rn(d) : 0.f;
}

// dst[r] += dis[r]*dis[c] * src[c]  over 800k edges; 4 dims per thread
__global__ void spmm_ui(const float* __restrict__ src, float* __restrict__ dst,
                        const int* __restrict__ rows, const int* __restrict__ cols,
                        const float* __restrict__ dis) {
  size_t gid = blockIdx.x * (size_t)blockDim.x + threadIdx.x;
  if (gid >= (size_t)NE2 * 16) return;
  int e = (int)(gid >> 4), q = (int)(gid & 15);
  int r = rows[e], c = cols[e];
  float w = dis[r] * dis[c];
  v4f v = *(const v4f*)(src + (size_t)c * DIM + q * 4);
  float* d = dst + (size_t)r * DIM + q * 4;
  atomicAdd(d + 0, w * v.x);
  atomicAdd(d + 1, w * v.y);
  atomicAdd(d + 2, w * v.z);
  atomicAdd(d + 3, w * v.w);
}

__global__ void finalize(const float* __restrict__ e0, const float* __restrict__ e1,
                         const float* __restrict__ e2, const float* __restrict__ hn,
                         float* __restrict__ out) {
  size_t i = blockIdx.x * (size_t)blockDim.x + threadIdx.x;
  if (i >= (size_t)NTOT * DIM) return;
  float v = (e0[i] + e1[i] + e2[i]) * (1.0f / 3.0f);
  if (i >= (size_t)NUSER * DIM) v += hn[i - (size_t)NUSER * DIM];
  out[i] = v;
}

// ---------------------------------------------------------------------------
extern "C" void kernel_launch(void* const* d_in, const int* in_sizes, int n_in,
                              void* d_out, int out_size, void* d_ws, size_t ws_size,
                              hipStream_t stream) {
  const float* user_emb = (const float*)d_in[0];
  const float* item_emb = (const float*)d_in[1];
  const float* v_feat   = (const float*)d_in[2];
  const float* t_feat   = (const float*)d_in[3];
  const float* W_v      = (const float*)d_in[4];
  const float* b_v      = (const float*)d_in[5];
  const float* W_t      = (const float*)d_in[6];
  const float* b_t      = (const float*)d_in[7];
  const float* Wq1      = (const float*)d_in[8];
  const float* bq1      = (const float*)d_in[9];
  const float* wq2      = (const float*)d_in[10];
  const float* orig_img_w = (const float*)d_in[11];
  const float* orig_txt_w = (const float*)d_in[12];
  const int*   edge_index = (const int*)d_in[13];
  const int*   orig_img_c = (const int*)d_in[14];
  const int*   orig_txt_c = (const int*)d_in[15];
  float* out = (float*)d_out;

  float* ws = (float*)d_ws;
  size_t o = 0;
  float* image_feats = ws + o; o += (size_t)NITEM * DIM;
  float* text_feats  = ws + o; o += (size_t)NITEM * DIM;
  float* xn_img      = ws + o; o += (size_t)NITEM * DIM;
  float* xn_txt      = ws + o; o += (size_t)NITEM * DIM;
  float* topv_img    = ws + o; o += (size_t)NITEM * TOPKN;
  float* topv_txt    = ws + o; o += (size_t)NITEM * TOPKN;
  int*   topc_img    = (int*)(ws + o); o += (size_t)NITEM * TOPKN;
  int*   topc_txt    = (int*)(ws + o); o += (size_t)NITEM * TOPKN;
  float* dis_img     = ws + o; o += NITEM;
  float* dis_txt     = ws + o; o += NITEM;
  float* image_item  = ws + o; o += (size_t)NITEM * DIM;
  float* text_item   = ws + o; o += (size_t)NITEM * DIM;
  float* h_norm      = ws + o; o += (size_t)NITEM * DIM;
  float* deg_ui      = ws + o; o += NTOT;
  float* dis_ui      = ws + o; o += NTOT;
  float* ego0        = ws + o; o += (size_t)NTOT * DIM;
  float* ego1        = ws + o; o += (size_t)NTOT * DIM;
  float* ego2        = ws + o; o += (size_t)NTOT * DIM;

  const int stripes = NITEM / 16;                 // 500
  const int wmma_blocks = (stripes + 7) / 8;      // 8 waves / 256-thread block

  // 1) feature projections (WMMA f32)
  proj_gemm_wmma<<<wmma_blocks, 256, 0, stream>>>(v_feat, W_v, b_v, image_feats, NITEM, VDIM);
  proj_gemm_wmma<<<wmma_blocks, 256, 0, stream>>>(t_feat, W_t, b_t, text_feats, NITEM, TDIM);

  // 2) row normalize
  row_normalize<<<(NITEM + 7) / 8, 256, 0, stream>>>(image_feats, xn_img, NITEM);
  row_normalize<<<(NITEM + 7) / 8, 256, 0, stream>>>(text_feats,  xn_txt, NITEM);

  // 3) fused sim + top-k (WMMA f32 + async LDS staging)
  sim_topk_wmma<<<wmma_blocks, 256, 0, stream>>>(xn_img, topv_img, topc_img);
  sim_topk_wmma<<<wmma_blocks, 256, 0, stream>>>(xn_txt, topv_txt, topc_txt);

  // 4) symmetric normalization factors
  compute_dis<<<(NITEM + 255) / 256, 256, 0, stream>>>(topv_img, dis_img, NITEM);
  compute_dis<<<(NITEM + 255) / 256, 256, 0, stream>>>(topv_txt, dis_txt, NITEM);

  // 5) fused 20-nnz/row SpMM against item_emb
  spmm_knn<<<(NITEM + 7) / 8, 256, 0, stream>>>(item_emb, topv_img, topc_img, dis_img,
                                                orig_img_w, orig_img_c, image_item);
  spmm_knn<<<(NITEM + 7) / 8, 256, 0, stream>>>(item_emb, topv_txt, topc_txt, dis_txt,
                                                orig_txt_w, orig_txt_c, text_item);

  // 6) attention gate + h normalization
  attention_h<<<200, 256, 0, stream>>>(image_item, text_item, Wq1, bq1, wq2, h_norm);

  // 7) UI LightGCN propagation
  build_ego<<<((size_t)NTOT * DIM + 255) / 256, 256, 0, stream>>>(user_emb, item_emb, ego0);
  zero_f32<<<512, 256, 0, stream>>>(deg_ui, NTOT);
  deg_count<<<(NE2 + 255) / 256, 256, 0, stream>>>(edge_index, deg_ui);
  dis_from_deg<<<(NTOT + 255) / 256, 256, 0, stream>>>(deg_ui, dis_ui, NTOT);

  const int* e_rows = edge_index;
  const int* e_cols = edge_index + NE2;
  size_t spmm_threads = (size_t)NE2 * 16;
  zero_f32<<<2048, 256, 0, stream>>>(ego1, (size_t)NTOT * DIM);
  spmm_ui<<<(spmm_threads + 255) / 256, 256, 0, stream>>>(ego0, ego1, e_rows, e_cols, dis_ui);
  zero_f32<<<2048, 256, 0, stream>>>(ego2, (size_t)NTOT * DIM);
  spmm_ui<<<(spmm_threads + 255) / 256, 256, 0, stream>>>(ego1, ego2, e_rows, e_cols, dis_ui);

  // 8) mean over layers + add normalized multimodal signal
  finalize<<<((size_t)NTOT * DIM + 255) / 256, 256, 0, stream>>>(ego0, ego1, ego2, h_norm, out);
}